// BilinearRoiPooling_45973329936534
// MI455X (gfx1250) — compile-verified
//
#include <hip/hip_runtime.h>
#include <cstddef>

// Problem constants (from reference setup_inputs): features (128,128,512) f32,
// boxes (1024,4) f32, pool 7x7, out (1024,7,7,512) f32.
#define HF 128
#define WF 128
#define CCH 512
#define PH 7
#define PW 7
#define NTHREADS 128          // 4 waves of 32
#define LDS_STRIDE 528        // 512 ch + 16 pad floats -> conflict-free B loads
#define LDS_ROWS 32           // 16 "top" (y0) rows + 16 "bottom" (y0+1) rows

typedef __attribute__((ext_vector_type(2))) float v2f;
typedef __attribute__((ext_vector_type(4))) float v4f;
typedef __attribute__((ext_vector_type(8))) float v8f;
typedef __attribute__((ext_vector_type(4))) int   v4i;

typedef __attribute__((address_space(1))) v4i gv4i;   // global 16B vector
typedef __attribute__((address_space(3))) v4i lv4i;   // LDS 16B vector

#if __has_builtin(__builtin_amdgcn_global_load_async_to_lds_b128)
#define HAVE_ASYNC 1
#else
#define HAVE_ASYNC 0
#endif

__device__ __forceinline__ void async_copy16(const float* gsrc, float* ldst) {
#if HAVE_ASYNC
  __builtin_amdgcn_global_load_async_to_lds_b128(
      (gv4i*)gsrc, (lv4i*)ldst, /*imm offset*/0, /*cpol*/0);
#else
  *(v4f*)ldst = *(const v4f*)gsrc;   // synchronous fallback
#endif
}

__device__ __forceinline__ void async_wait_all() {
#if HAVE_ASYNC
#if __has_builtin(__builtin_amdgcn_s_wait_asynccnt)
  __builtin_amdgcn_s_wait_asynccnt(0);
#else
  asm volatile("s_wait_asynccnt 0" ::: "memory");
#endif
#endif
}

__launch_bounds__(NTHREADS)
__global__ void BilinearRoiPooling_kernel(const float* __restrict__ feat,
                                          const float* __restrict__ boxes,
                                          const int* __restrict__ ihp,
                                          const int* __restrict__ iwp,
                                          float* __restrict__ out) {
  __shared__ float lds[LDS_ROWS * LDS_STRIDE];

  const int blk  = blockIdx.x;        // = box*7 + py
  const int b    = blk / PH;
  const int py   = blk - b * PH;
  const int tid  = threadIdx.x;
  const int lane = tid & 31;
  const int wave = tid >> 5;
  const int half = lane >> 4;         // 0: lanes 0-15, 1: lanes 16-31
  const int n    = lane & 15;         // channel-within-tile / A-row id

  const float H = (float)ihp[0];
  const float W = (float)iwp[0];

  const float xc = boxes[b * 4 + 0];
  const float yc = boxes[b * 4 + 1];
  const float bw = boxes[b * 4 + 2];
  const float bh = boxes[b * 4 + 3];

  // ---- y sampling for this pooled row (shared by all 7 samples) ----
  const float ysv = (float)py * (2.0f / 6.0f) - 1.0f;
  const float g0  = ysv * ((bh - 1.0f) / (H - 1.0f)) +
                    (2.0f * yc - H - 1.0f) / (H - 1.0f);
  const float yf  = (g0 + 1.0f) * 0.5f * (float)(HF - 1);
  const float y0f = floorf(yf);
  const float wy  = yf - y0f;
  const bool  vy0 = (y0f >= 0.0f) && (y0f <= (float)(HF - 1));
  const bool  vy1 = (y0f + 1.0f >= 0.0f) && (y0f + 1.0f <= (float)(HF - 1));
  const int   y0c = min(max((int)y0f, 0), HF - 1);
  const int   y1c = min(max((int)y0f + 1, 0), HF - 1);

  auto xf_of = [&](int p) -> float {
    float xsv = (float)p * (2.0f / 6.0f) - 1.0f;
    float g1  = xsv * ((bw - 1.0f) / (W - 1.0f)) +
                (2.0f * xc - W - 1.0f) / (W - 1.0f);
    return (g1 + 1.0f) * 0.5f * (float)(WF - 1);
  };

  // ---- zero K-pad rows 14,15 (top) and 14,15 (bottom): A cols there are 0,
  // but 0 * garbage(NaN) would poison the accumulator ----
  for (int i = tid; i < 4 * CCH; i += NTHREADS) {
    int rsel = i >> 9;                 // 0..3
    int colc = i & (CCH - 1);
    int row  = (rsel < 2) ? (14 + rsel) : (16 + 14 + (rsel - 2));
    lds[row * LDS_STRIDE + colc] = 0.0f;
  }

  // ---- async-stage 2 y-rows x 14 x-columns x 512 ch into LDS (57 KB) ----
  // job: slot = which (y,xcol) row [0..27], chunk = which float4 [0..127]
  for (int j = tid; j < 28 * NTHREADS; j += NTHREADS) {
    int chunk = j & 127;
    int slot  = j >> 7;                // 0..27
    int ysel  = (slot >= 14) ? 1 : 0;
    int cidx  = slot - 14 * ysel;      // 0..13 -> column slot
    int p     = cidx >> 1;             // pooled col 0..6
    int isx1  = cidx & 1;              // 0: x0, 1: x0+1
    float xfv = xf_of(p);
    int   x0  = (int)floorf(xfv);
    int   xi  = min(max(x0 + isx1, 0), WF - 1);
    int   row = ysel ? y1c : y0c;
    const float* gsrc = feat + ((size_t)(row * WF + xi) * CCH) + (size_t)chunk * 4;
    float* ldst = &lds[(ysel * 16 + cidx) * LDS_STRIDE + chunk * 4];
    async_copy16(gsrc, ldst);
  }
  async_wait_all();
  __syncthreads();

  // ---- per-lane A-matrix weights (row m = lane&15 = pooled col), masks and
  // wy folded in so masked OOB corners contribute exactly 0 ----
  float wt0 = 0.0f, wt1 = 0.0f, wb0 = 0.0f, wb1 = 0.0f;
  if (n < PW) {
    float xfv = xf_of(n);
    float x0f = floorf(xfv);
    float wx  = xfv - x0f;
    bool  vx0 = (x0f >= 0.0f) && (x0f <= (float)(WF - 1));
    bool  vx1 = (x0f + 1.0f >= 0.0f) && (x0f + 1.0f <= (float)(WF - 1));
    float u = 1.0f - wy;
    wt0 = u  * (1.0f - wx) * ((vy0 && vx0) ? 1.0f : 0.0f);
    wt1 = u  * wx          * ((vy0 && vx1) ? 1.0f : 0.0f);
    wb0 = wy * (1.0f - wx) * ((vy1 && vx0) ? 1.0f : 0.0f);
    wb1 = wy * wx          * ((vy1 && vx1) ? 1.0f : 0.0f);
  }
  // A layout (f32 16x4 chunk c of K=16): VGPR0 lane l<16 -> A[l,4c], l>=16 ->
  // A[l-16,4c+2]; VGPR1 -> A[.,4c+1]/A[.,4c+3]. Row m has nonzeros only at
  // cols {2m, 2m+1}, so lane pair is {w0,w1} iff m == 2c + half, else 0.
  v2f At[4], Ab[4];
#pragma unroll
  for (int c = 0; c < 4; ++c) {
    bool sel = (n == 2 * c + half);
    v2f at; at.x = sel ? wt0 : 0.0f; at.y = sel ? wt1 : 0.0f;
    v2f ab; ab.x = sel ? wb0 : 0.0f; ab.y = sel ? wb1 : 0.0f;
    At[c] = at; Ab[c] = ab;
  }

  // ---- WMMA: per wave, 8 channel tiles of 16; D = Atop x Btop + Abot x Bbot ----
  float* outbase = out + (size_t)blk * PW * CCH;   // + M*512 + ch
  for (int nt = 0; nt < 8; ++nt) {
    const int ch = wave * 128 + nt * 16 + n;
    v8f acc = {};
#pragma unroll
    for (int c = 0; c < 4; ++c) {              // top rows (y0): K rows 4c..4c+3
      int r0 = 4 * c + 2 * half;
      v2f Bt; Bt.x = lds[(r0)     * LDS_STRIDE + ch];
              Bt.y = lds[(r0 + 1) * LDS_STRIDE + ch];
      acc = __builtin_amdgcn_wmma_f32_16x16x4_f32(false, At[c], false, Bt,
                                                  (short)0, acc, false, false);
    }
#pragma unroll
    for (int c = 0; c < 4; ++c) {              // bottom rows (y0+1)
      int r0 = 16 + 4 * c + 2 * half;
      v2f Bb; Bb.x = lds[(r0)     * LDS_STRIDE + ch];
              Bb.y = lds[(r0 + 1) * LDS_STRIDE + ch];
      acc = __builtin_amdgcn_wmma_f32_16x16x4_f32(false, Ab[c], false, Bb,
                                                  (short)0, acc, false, false);
    }
    // D layout: VGPR r, lanes 0-15 -> M=r, lanes 16-31 -> M=r+8. Only M<7 valid.
    if (half == 0) {
#pragma unroll
      for (int r = 0; r < 7; ++r) {
        outbase[(size_t)r * CCH + wave * 128 + nt * 16 + n] = acc[r];
      }
    }
  }
}

extern "C" void kernel_launch(void* const* d_in, const int* in_sizes, int n_in,
                              void* d_out, int out_size, void* d_ws, size_t ws_size,
                              hipStream_t stream) {
  const float* feat  = (const float*)d_in[0];
  const float* boxes = (const float*)d_in[1];
  const int*   ih    = (const int*)d_in[2];
  const int*   iw    = (const int*)d_in[3];
  float*       out   = (float*)d_out;
  const int nboxes = in_sizes[1] / 4;          // (B,4) boxes
  dim3 grid(nboxes * PH);                      // one block per (box, pooled row)
  BilinearRoiPooling_kernel<<<grid, NTHREADS, 0, stream>>>(feat, boxes, ih, iw, out);
}